// Prediction_17386027614913
// MI455X (gfx1250) — compile-verified
//
#include <hip/hip_runtime.h>
#include <stdint.h>

// ---- constants mirroring the reference ----
#define INP_F       608.0f
#define NUM_CLASSES 20
#define OBJ_T       0.6f
#define CONF_T      0.05f
#define LOC_T       0.5f
#define NMS_T       0.1f
#define SIGMA_NMS   0.5f
#define NEGV        (-1e9f)
#define BATCH       64
#define NTOT        22743     // 3*19*19 + 3*38*38 + 3*76*76
#define NPAD        22784     // padded LDS score array length
#define MCAND       256
#define TOPK        8

__device__ __forceinline__ float sigmoidf_(float x) { return 1.0f / (1.0f + __expf(-x)); }

// ----------------------------------------------------------------------------
// Pass 1: streaming decode (HBM-bound: one pass over 151 MB of predictions)
// ----------------------------------------------------------------------------
__global__ void decode_kernel(const float* __restrict__ pred, int G, float stride,
                              float aw0, float ah0, float aw1, float ah1,
                              float aw2, float ah2,
                              int n_off, int n_lvl,
                              float* __restrict__ scores,
                              float4* __restrict__ boxes,
                              int* __restrict__ clsid)
{
    const int total = BATCH * n_lvl;
    for (int gid = blockIdx.x * blockDim.x + threadIdx.x; gid < total;
         gid += gridDim.x * blockDim.x) {
        int b    = gid / n_lvl;
        int r    = gid - b * n_lvl;
        int gg   = G * G;
        int a    = r / gg;
        int cell = r - a * gg;
        int gy   = cell / G;
        int gx   = cell - gy * G;

        const float* p = pred + ((((size_t)b * 3 + a) * G + gy) * G + gx) * 26;
        __builtin_prefetch(p, 0, 0);   // global_prefetch_b8 on gfx1250

        float aw = (a == 0) ? aw0 : ((a == 1) ? aw1 : aw2);
        float ah = (a == 0) ? ah0 : ((a == 1) ? ah1 : ah2);

        // ALPHA == 1.0 -> xy = sigmoid + grid offset
        float cx  = (sigmoidf_(p[0]) + (float)gx) * stride;
        float cy  = (sigmoidf_(p[1]) + (float)gy) * stride;
        float w   = __expf(p[2]) * aw;   // (exp * anchor/stride) * stride
        float h   = __expf(p[3]) * ah;
        float obj = sigmoidf_(p[4]);
        float loc = sigmoidf_(p[5]);

        // softmax max/argmax over 20 classes: conf = 1/sum(exp(l - max))
        float m = p[6]; int bc = 0;
        #pragma unroll
        for (int c = 1; c < NUM_CLASSES; ++c) {
            float v = p[6 + c];
            if (v > m) { m = v; bc = c; }
        }
        float sum = 0.0f;
        #pragma unroll
        for (int c = 0; c < NUM_CLASSES; ++c) sum += __expf(p[6 + c] - m);
        float conf = 1.0f / sum;

        float x1 = fminf(fmaxf(cx - 0.5f * w, 0.0f), INP_F);
        float y1 = fminf(fmaxf(cy - 0.5f * h, 0.0f), INP_F);
        float x2 = fminf(fmaxf(cx + 0.5f * w, 0.0f), INP_F);
        float y2 = fminf(fmaxf(cy + 0.5f * h, 0.0f), INP_F);

        float oc   = obj * conf;
        float area = (x2 - x1) * (y2 - y1);
        bool valid = (obj >= OBJ_T) && (area >= 0.0f) && (oc >= CONF_T) && (loc >= LOC_T);
        // BETA = 0.5: (obj*conf)^0.5 * loc^0.5
        float score = __fsqrt_rn(oc) * __fsqrt_rn(loc);

        size_t o  = (size_t)b * NTOT + n_off + r;
        scores[o] = valid ? score : NEGV;
        boxes[o]  = make_float4(x1, y1, x2, y2);
        clsid[o]  = bc;
    }
}

// ----------------------------------------------------------------------------
// Pass 2: per-image exact top-256. CDNA5 path: the whole 89 KB score vector is
// staged into LDS with global_load_async_to_lds_b32 (ASYNCcnt), then 256
// argmax rounds run entirely out of LDS (320 KB/WGP makes this possible).
// ----------------------------------------------------------------------------
__global__ __launch_bounds__(1024)
void topk_kernel(const float* __restrict__ scores, const float4* __restrict__ boxes,
                 const int* __restrict__ clsid,
                 float* __restrict__ cand_s, float4* __restrict__ cand_b,
                 int* __restrict__ cand_c)
{
    __shared__ float s_sc[NPAD];     // ~89 KB: full per-image score vector
    __shared__ float rv[1024];
    __shared__ int   ri[1024];
    __shared__ int   topi[MCAND];
    __shared__ float topv[MCAND];

    const int b   = blockIdx.x;
    const int tid = threadIdx.x;
    const float* src = scores + (size_t)b * NTOT;
    const unsigned long long sbase = (unsigned long long)(uintptr_t)src;

    // async global -> LDS stage of the score vector
    for (int i = tid; i < NPAD; i += 1024) {
        if (i < NTOT) {
            uint32_t lds = (uint32_t)(uintptr_t)&s_sc[i];   // LDS byte address (low 32 bits)
            uint32_t off = (uint32_t)(i * 4);
            asm volatile("global_load_async_to_lds_b32 %0, %1, %2"
                         :: "v"(lds), "v"(off), "s"(sbase) : "memory");
        } else {
            s_sc[i] = NEGV;                                 // padding
        }
    }
    asm volatile("s_wait_asynccnt 0" ::: "memory");
    __syncthreads();

    // 256 rounds of exact argmax (lowest-index tie-break, matching jnp.argmax)
    for (int round = 0; round < MCAND; ++round) {
        float best = -3.4e38f; int bi = tid;
        for (int i = tid; i < NPAD; i += 1024) {
            float v = s_sc[i];
            if (v > best) { best = v; bi = i; }
        }
        rv[tid] = best; ri[tid] = bi;
        __syncthreads();
        for (int off = 512; off > 0; off >>= 1) {
            if (tid < off) {
                float vo = rv[tid + off]; int io = ri[tid + off];
                if (vo > rv[tid] || (vo == rv[tid] && io < ri[tid])) { rv[tid] = vo; ri[tid] = io; }
            }
            __syncthreads();
        }
        if (tid == 0) {
            topi[round] = ri[0];
            topv[round] = rv[0];
            s_sc[ri[0]] = NEGV;      // remove winner
        }
        __syncthreads();
    }

    // gather candidate boxes/classes/scores
    if (tid < MCAND) {
        int   i = topi[tid];
        float v = topv[tid];
        size_t o  = (size_t)b * MCAND + tid;
        cand_s[o] = (v > NEGV * 0.5f) ? v : NEGV;
        cand_b[o] = boxes[(size_t)b * NTOT + i];
        cand_c[o] = clsid[(size_t)b * NTOT + i];
    }
}

// ----------------------------------------------------------------------------
// Pass 3: per-image sequential Gaussian soft-NMS (+1 pixel IoU convention),
// then top-8 by original score among kept. One 256-thread block per image.
// ----------------------------------------------------------------------------
__global__ __launch_bounds__(256)
void nms_kernel(const float* __restrict__ cand_s, const float4* __restrict__ cand_b,
                const int* __restrict__ cand_c, float* __restrict__ out)
{
    __shared__ float rv[MCAND];
    __shared__ int   ri[MCAND];
    __shared__ float sbx[4];
    __shared__ int   scj;

    const int b = blockIdx.x;
    const int t = threadIdx.x;
    const size_t o0 = (size_t)b * MCAND + t;

    float4 box = cand_b[o0];
    int    cls = cand_c[o0];
    float  s   = cand_s[o0];
    float  s0  = s;               // original (pre-decay) score for final ranking
    bool   kept = false;

    for (int it = 0; it < MCAND; ++it) {
        rv[t] = s; ri[t] = t;
        __syncthreads();
        for (int off = 128; off > 0; off >>= 1) {
            if (t < off) {
                float vo = rv[t + off]; int io = ri[t + off];
                if (vo > rv[t] || (vo == rv[t] && io < ri[t])) { rv[t] = vo; ri[t] = io; }
            }
            __syncthreads();
        }
        int j = ri[0]; float vmax = rv[0];
        __syncthreads();
        if (vmax < NMS_T) break;   // uniform: remaining reference iterations are no-ops

        if (t == j) { sbx[0] = box.x; sbx[1] = box.y; sbx[2] = box.z; sbx[3] = box.w;
                      scj = cls; kept = true; }
        __syncthreads();

        float ix1 = fmaxf(sbx[0], box.x), iy1 = fmaxf(sbx[1], box.y);
        float ix2 = fminf(sbx[2], box.z), iy2 = fminf(sbx[3], box.w);
        float inter = fmaxf(ix2 - ix1 + 1.0f, 0.0f) * fmaxf(iy2 - iy1 + 1.0f, 0.0f);
        float a1 = (sbx[2] - sbx[0] + 1.0f) * (sbx[3] - sbx[1] + 1.0f);
        float a2 = (box.z - box.x + 1.0f) * (box.w - box.y + 1.0f);
        float iou = inter / (a1 + a2 - inter + 1e-16f);
        float decay = __expf(-(iou * iou) / SIGMA_NMS);
        if (cls == scj) s *= decay;
        if (t == j)     s = NEGV;
        __syncthreads();
    }
    __syncthreads();

    // top-8 among kept, ranked by original score
    float f = kept ? s0 : NEGV;
    for (int k = 0; k < TOPK; ++k) {
        rv[t] = f; ri[t] = t;
        __syncthreads();
        for (int off = 128; off > 0; off >>= 1) {
            if (t < off) {
                float vo = rv[t + off]; int io = ri[t + off];
                if (vo > rv[t] || (vo == rv[t] && io < ri[t])) { rv[t] = vo; ri[t] = io; }
            }
            __syncthreads();
        }
        int j = ri[0]; float val = rv[0];
        __syncthreads();
        if (t == j) {
            float* o = out + ((size_t)b * TOPK + k) * 6;
            bool ok = val > NEGV * 0.5f;
            o[0] = ok ? box.x : 0.0f;
            o[1] = ok ? box.y : 0.0f;
            o[2] = ok ? box.z : 0.0f;
            o[3] = ok ? box.w : 0.0f;
            o[4] = ok ? val   : 0.0f;
            o[5] = ok ? (float)cls : 0.0f;
            f = NEGV;
        }
        __syncthreads();
    }
}

// ----------------------------------------------------------------------------
extern "C" void kernel_launch(void* const* d_in, const int* in_sizes, int n_in,
                              void* d_out, int out_size, void* d_ws, size_t ws_size,
                              hipStream_t stream)
{
    (void)in_sizes; (void)n_in; (void)out_size; (void)ws_size;
    const float* p0 = (const float*)d_in[0];   // (64,3,19,19,26)
    const float* p1 = (const float*)d_in[1];   // (64,3,38,38,26)
    const float* p2 = (const float*)d_in[2];   // (64,3,76,76,26)

    char* ws = (char*)d_ws;
    size_t off = 0;
    float*  scores = (float*) (ws + off); off += (size_t)BATCH * NTOT  * sizeof(float);
    float4* boxes  = (float4*)(ws + off); off += (size_t)BATCH * NTOT  * sizeof(float4);
    int*    clsid  = (int*)   (ws + off); off += (size_t)BATCH * NTOT  * sizeof(int);
    float*  cand_s = (float*) (ws + off); off += (size_t)BATCH * MCAND * sizeof(float);
    float4* cand_b = (float4*)(ws + off); off += (size_t)BATCH * MCAND * sizeof(float4);
    int*    cand_c = (int*)   (ws + off);

    // Level 0: G=19, stride=32, anchors (116,90) (156,198) (373,326)
    {
        int n_lvl = 3 * 19 * 19, total = BATCH * n_lvl;
        decode_kernel<<<(total + 255) / 256, 256, 0, stream>>>(
            p0, 19, 32.0f, 116.f, 90.f, 156.f, 198.f, 373.f, 326.f,
            0, n_lvl, scores, boxes, clsid);
    }
    // Level 1: G=38, stride=16, anchors (30,61) (62,45) (59,119)
    {
        int n_lvl = 3 * 38 * 38, total = BATCH * n_lvl;
        decode_kernel<<<(total + 255) / 256, 256, 0, stream>>>(
            p1, 38, 16.0f, 30.f, 61.f, 62.f, 45.f, 59.f, 119.f,
            1083, n_lvl, scores, boxes, clsid);
    }
    // Level 2: G=76, stride=8, anchors (10,13) (16,30) (33,23)
    {
        int n_lvl = 3 * 76 * 76, total = BATCH * n_lvl;
        decode_kernel<<<(total + 255) / 256, 256, 0, stream>>>(
            p2, 76, 8.0f, 10.f, 13.f, 16.f, 30.f, 33.f, 23.f,
            5415, n_lvl, scores, boxes, clsid);
    }

    topk_kernel<<<BATCH, 1024, 0, stream>>>(scores, boxes, clsid, cand_s, cand_b, cand_c);
    nms_kernel <<<BATCH,  256, 0, stream>>>(cand_s, cand_b, cand_c, (float*)d_out);
}